// MRA_36833639530629
// MI455X (gfx1250) — compile-verified
//
#include <hip/hip_runtime.h>
#include <stdint.h>

// Problem geometry (fixed by the reference)
#define HH 192
#define WW 192
#define AH 64
#define AW 64
#define PLANE  (HH * WW)   // 36864 elements per (b,c) plane
#define APLANE (AH * AW)   // 4096 pooled elements
#define NCH 256
#define NPLANES 2048       // B * C = 8 * 256

typedef unsigned int u32x4 __attribute__((ext_vector_type(4)));
typedef unsigned int u32x8 __attribute__((ext_vector_type(8)));

// ---------------------------------------------------------------------------
// Tensor Data Mover: DMA `tile_rows` x 192 fp32 elements from global memory
// into LDS at byte offset `lds_byte_off`.  D# layout per CDNA5 ISA ch.8:
//   group0: [1:0]=count=1, [63:32]=lds_addr, [120:64]=global_addr, [127:126]=2
//   group1: [17:16]=data_size(4B), [79:48]=tensor_dim0, [111:80]=tensor_dim1,
//           [127:112]=tile_dim0, [143:128]=tile_dim1, [207:160]=dim0_stride
// 2-D tile -> D# groups 2/3 are NULL (2-operand tensor_load_to_lds form).
// ---------------------------------------------------------------------------
__device__ __forceinline__ void tdm_load_rows_to_lds(unsigned lds_byte_off,
                                                     const float* gsrc,
                                                     unsigned tile_rows) {
  unsigned long long ga = (unsigned long long)(uintptr_t)gsrc;
  u32x4 g0;
  g0.x = 1u;                                               // count=1 (user D#)
  g0.y = lds_byte_off;                                     // LDS byte address
  g0.z = (unsigned)ga;                                     // global_addr[31:0]
  g0.w = (unsigned)((ga >> 32) & 0x01FFFFFFu) | (2u << 30); // addr[56:32]|type=2
  u32x8 g1;
  g1[0] = (2u << 16);                                      // data_size = 4B
  g1[1] = ((unsigned)WW & 0xFFFFu) << 16;                  // tensor_dim0 lo16
  g1[2] = ((unsigned)HH & 0xFFFFu) << 16;                  // dim0 hi=0 | dim1 lo16
  g1[3] = ((unsigned)WW & 0xFFFFu) << 16;                  // dim1 hi=0 | tile_dim0
  g1[4] = tile_rows & 0xFFFFu;                             // tile_dim1 | tile_dim2=0
  g1[5] = (unsigned)WW;                                    // tensor_dim0_stride
  g1[6] = 0u;
  g1[7] = 0u;
  asm volatile("tensor_load_to_lds %0, %1" : : "s"(g0), "s"(g1) : "memory");
}

__device__ __forceinline__ float stval(const float* st, int y, int x) {
  return ((unsigned)y < (unsigned)AH && (unsigned)x < (unsigned)AW)
             ? st[y * AW + x] : 0.f;
}

__global__ __launch_bounds__(256) void mra_fused(
    const float* __restrict__ x,
    const float* __restrict__ wh1, const float* __restrict__ wv1,
    const float* __restrict__ wh2, const float* __restrict__ wv2,
    const float* __restrict__ gma, const float* __restrict__ bta,
    const float* __restrict__ mu,  const float* __restrict__ var,
    float* __restrict__ out)
{
  // 163,840 bytes total: full x plane + pooled plane live in LDS.
  __shared__ float smem[PLANE + APLANE];
  float* sx = smem;            // 192x192 input plane
  float* st = smem + PLANE;    // 64x64 pooled plane

  const int tid   = threadIdx.x;
  const int plane = blockIdx.x;            // b*256 + c
  const int c     = plane & (NCH - 1);
  const float* xp = x + (size_t)plane * PLANE;
  float* op       = out + (size_t)plane * PLANE;

  // ---------------- Phase A: TDM DMA plane -> LDS (8 waves x 24 rows) -------
  {
    unsigned wave = (unsigned)tid >> 5;
    wave = __builtin_amdgcn_readfirstlane(wave);       // provably wave-uniform
    const unsigned rows = HH / 8;                      // 24
    tdm_load_rows_to_lds(wave * rows * WW * 4u,
                         xp + (size_t)wave * rows * WW, rows);
    __builtin_amdgcn_s_wait_tensorcnt(0);
  }
  __syncthreads();

  // ---------------- Phase B: maxpool3x3(SAME) + blurpool4x4(stride3) --------
  // Separable max: 3x3 max = vert-max of horiz-max over a clamped 6x6 region.
  // Reflect pad [1,2] only ever produces coord -1 -> 1 (upper never occurs),
  // and that reflected tap equals the fx=2 / fy=2 tap, fixed with a select.
  // SAME maxpool == clamped window (clamped dup indices leave the max intact).
  {
    const float bw4[4] = {0.125f, 0.375f, 0.375f, 0.125f};
#pragma unroll 1
    for (int k = 0; k < 16; ++k) {
      int idx = tid + 256 * k;
      int oh = idx >> 6, ow = idx & 63;
      int row0 = 3 * oh - 2;
      int col0 = 3 * ow - 2;

      float v[6][6];                       // clamped 6x6 input region
#pragma unroll
      for (int t = 0; t < 6; ++t) {
        int y = row0 + t; y = y < 0 ? 0 : (y > HH - 1 ? HH - 1 : y);
        const float* r = sx + y * WW;
#pragma unroll
        for (int u = 0; u < 6; ++u) {
          int xx = col0 + u; xx = xx < 0 ? 0 : (xx > WW - 1 ? WW - 1 : xx);
          v[t][u] = r[xx];
        }
      }

      float hm[6][4];                      // horizontal 3-max at 4 blur cols
#pragma unroll
      for (int t = 0; t < 6; ++t)
#pragma unroll
        for (int fx = 0; fx < 4; ++fx)
          hm[t][fx] = fmaxf(fmaxf(v[t][fx], v[t][fx + 1]), v[t][fx + 2]);
      if (ow == 0) {                       // reflected tap p=-1 -> p=1
#pragma unroll
        for (int t = 0; t < 6; ++t) hm[t][0] = hm[t][2];
      }

      float mp[4][4];                      // vertical 3-max -> maxpool values
#pragma unroll
      for (int fy = 0; fy < 4; ++fy)
#pragma unroll
        for (int fx = 0; fx < 4; ++fx)
          mp[fy][fx] = fmaxf(fmaxf(hm[fy][fx], hm[fy + 1][fx]), hm[fy + 2][fx]);
      if (oh == 0) {                       // reflected tap q=-1 -> q=1
#pragma unroll
        for (int fx = 0; fx < 4; ++fx) mp[0][fx] = mp[2][fx];
      }

      float acc = 0.f;                     // binomial 4x4 blur
#pragma unroll
      for (int fy = 0; fy < 4; ++fy)
#pragma unroll
        for (int fx = 0; fx < 4; ++fx)
          acc += mp[fy][fx] * (bw4[fy] * bw4[fx]);
      st[idx] = acc;
    }
  }
  __syncthreads();

  // ---------------- Phase C+D: 4 depthwise convs + BN + sigmoid + x3 gate ---
  // Diagonal branches reduce (via h/v transform algebra) to:
  //   x_h2[i,j] = sum st[i+di, j-di+dk] * wh2[di+5, dk+1]
  //   x_w2[i,j] = sum st[i+dk-dj, j+dj] * wv2[dk+1, dj+5]
  {
    const float inv   = gma[c] * rsqrtf(var[c] + 1e-5f);
    const float shift = bta[c] - mu[c] * inv;
    const int i0 = tid >> 6;   // base gate row (0..3), rows i0+4k
    const int j  = tid & 63;   // fixed gate column

    float acc[16];
#pragma unroll
    for (int k = 0; k < 16; ++k) acc[k] = 0.f;

    const float* wh1c = wh1 + c * 33;
    const float* wh2c = wh2 + c * 33;
    const float* wv1c = wv1 + c * 33;
    const float* wv2c = wv2 + c * 33;

    // (11x3) kernels: horizontal + first diagonal
#pragma unroll 1
    for (int di = -5; di <= 5; ++di) {
#pragma unroll 1
      for (int dk = -1; dk <= 1; ++dk) {
        float w1 = wh1c[(di + 5) * 3 + (dk + 1)];   // uniform scalar loads
        float w2 = wh2c[(di + 5) * 3 + (dk + 1)];
        int c1 = j + dk;
        int c2 = j - di + dk;
#pragma unroll
        for (int k = 0; k < 16; ++k) {
          int y = i0 + 4 * k + di;
          acc[k] += stval(st, y, c1) * w1 + stval(st, y, c2) * w2;
        }
      }
    }
    // (3x11) kernels: vertical + second diagonal
#pragma unroll 1
    for (int dk = -1; dk <= 1; ++dk) {
#pragma unroll 1
      for (int dj = -5; dj <= 5; ++dj) {
        float w1 = wv1c[(dk + 1) * 11 + (dj + 5)];
        float w2 = wv2c[(dk + 1) * 11 + (dj + 5)];
        int cc = j + dj;
#pragma unroll
        for (int k = 0; k < 16; ++k) {
          int y1 = i0 + 4 * k + dk;
          acc[k] += stval(st, y1, cc) * w1 + stval(st, y1 - dj, cc) * w2;
        }
      }
    }

    // BN + sigmoid, then scale the 3x3 output block (x re-read from LDS).
#pragma unroll 1
    for (int k = 0; k < 16; ++k) {
      float att = acc[k] * inv + shift;
      float g = 1.f / (1.f + __expf(-att));
      int ybase = 3 * (i0 + 4 * k);
      int xbase = 3 * j;
#pragma unroll
      for (int r = 0; r < 3; ++r) {
        const float* srow = sx + (ybase + r) * WW + xbase;
        float* orow = op + (size_t)(ybase + r) * WW + xbase;
        orow[0] = srow[0] * g;
        orow[1] = srow[1] * g;
        orow[2] = srow[2] * g;
      }
    }
  }
}

extern "C" void kernel_launch(void* const* d_in, const int* in_sizes, int n_in,
                              void* d_out, int out_size, void* d_ws, size_t ws_size,
                              hipStream_t stream) {
  (void)in_sizes; (void)n_in; (void)out_size; (void)d_ws; (void)ws_size;
  const float* x   = (const float*)d_in[0];
  const float* wh1 = (const float*)d_in[1];
  const float* wv1 = (const float*)d_in[2];
  const float* wh2 = (const float*)d_in[3];
  const float* wv2 = (const float*)d_in[4];
  const float* gma = (const float*)d_in[5];
  const float* bta = (const float*)d_in[6];
  const float* mu  = (const float*)d_in[7];
  const float* var = (const float*)d_in[8];
  mra_fused<<<dim3(NPLANES), dim3(256), 0, stream>>>(
      x, wh1, wv1, wh2, wv2, gma, bta, mu, var, (float*)d_out);
}